// CvxNet_59055800320739
// MI455X (gfx1250) — compile-verified
//
#include <hip/hip_runtime.h>
#include <hip/hip_bf16.h>

// ---------------------------------------------------------------------------
// Problem constants (from reference)
// ---------------------------------------------------------------------------
#define NI 16
#define NO 8
#define NU 240
#define NH 30          // horizon N
#define NK 480         // N*NI
#define NINEQ 1440     // 2*(NU + NI*N)
#define QP_ITERS 25
#define SIGMA 0.1f
#define BATCH 256
#define NT 256         // threads per block (8 wave32)

// Workspace layout (float offsets)
#define OFF_AHAT 0                    // [480][16]
#define OFF_BHAT 7680                 // [480][240]
#define OFF_QB   122880               // [480][240]  Q_diag @ B_hat
#define OFF_QM   238080               // [16][16]
#define OFF_RM   238336               // [8][8]
#define OFF_QHAT 238400               // [240][240]
#define OFF_TWOQ 296000               // [240][240]
#define OFF_P    353600               // [B][240]
#define OFF_H    415040               // [B][1440]
#define OFF_BC   783680               // [B]  (b+c cost terms)

// PDIP kernel dynamic LDS layout (float offsets)
#define LM_M    0                     // 240*241 padded KKT matrix
#define LM_Z    57840
#define LM_RHS  58080
#define LM_DD   58320
#define LM_BZ   58560                 // 480 scratch (Bz / w / Bdz)
#define LM_SQE  59040                 // 480
#define LM_LAM  59520                 // 1440
#define LM_S    60960
#define LM_H    62400
#define LM_RP   63840
#define LM_T    65280
#define LM_DS   66720
#define LM_DL   68160
#define LM_RED  69600                 // 256
#define PDIP_SMEM_FLOATS 69856
#define PDIP_SMEM_BYTES (PDIP_SMEM_FLOATS * 4)

typedef __attribute__((ext_vector_type(2))) float v2f;
typedef __attribute__((ext_vector_type(8))) float v8f;

// ---------------------------------------------------------------------------
// Wave-level f32 SYRK row-chunk: for one 16-row tile-row (ti) compute 5
// adjacent 16x16 output tiles, D_u += sum_k W[k,i]*W2[k, j_u], k=0..479.
// A-fragment is loaded ONCE per k-step and reused for all 5 column tiles
// (2 + 10 loads per 5 WMMAs instead of 4 per WMMA -> ~40% less L2 traffic).
// Ac points at A-column (ti*16+lm); Bc at first B-column of the chunk
// (chunk*80+lm); both in a [480,240] row-major matrix. Optional symmetric
// scale (sqrt(e), LDS) applied to both operands.
// VGPR layouts per CDNA5 ISA 7.12.2 (16x4 fp32 A; 16x16 fp32 C/D).
// ---------------------------------------------------------------------------
__device__ __forceinline__ void wmma_syrk480_row5(const float* Ac,
                                                  const float* Bc,
                                                  const float* scale,
                                                  int half, v8f acc[5]) {
    for (int kb = 0; kb < NK; kb += 4) {
        const int k0 = kb + 2 * half;
        const float s0v = scale ? scale[k0]     : 1.0f;
        const float s1v = scale ? scale[k0 + 1] : 1.0f;
        v2f a;
        a.x = s0v * Ac[k0 * NU];
        a.y = s1v * Ac[(k0 + 1) * NU];
#pragma unroll
        for (int u = 0; u < 5; ++u) {
            v2f b;
            b.x = s0v * Bc[u * 16 + k0 * NU];
            b.y = s1v * Bc[u * 16 + (k0 + 1) * NU];
            acc[u] = __builtin_amdgcn_wmma_f32_16x16x4_f32(
                false, a, false, b, (short)0, acc[u], false, false);
        }
    }
}

// ---------------------------------------------------------------------------
// Kernel 0: build A_hat, B_hat, Qm, Rm, QB (= blockdiag(Qm) @ B_hat).
// Single workgroup; all intermediates held in LDS to avoid intra-kernel
// global RAW hazards.
// ---------------------------------------------------------------------------
__global__ void __launch_bounds__(NT) k_precompute(const float* __restrict__ Q,
                                                   const float* __restrict__ R,
                                                   const float* __restrict__ A,
                                                   const float* __restrict__ B,
                                                   float* __restrict__ ws) {
    __shared__ float As[256];        // A
    __shared__ float Bs[128];        // B
    __shared__ float Apows[30 * 256];// A^1..A^30
    __shared__ float ABs[30 * 128];  // A^i B
    __shared__ float Qms[256];       // Q Q^T
    const int tid = threadIdx.x;
    float* Ahat = ws + OFF_AHAT;
    float* Bhat = ws + OFF_BHAT;
    float* QB   = ws + OFF_QB;
    float* Qmg  = ws + OFF_QM;
    float* Rmg  = ws + OFF_RM;

    As[tid] = A[tid];
    if (tid < 128) Bs[tid] = B[tid];

    { // Qm = Q @ Q^T
        const int r = tid >> 4, c = tid & 15;
        float acc = 0.f;
        for (int t = 0; t < 16; ++t) acc += Q[r * 16 + t] * Q[c * 16 + t];
        Qms[tid] = acc;
        Qmg[tid] = acc;
    }
    if (tid < 64) { // Rm = R @ R^T
        const int r = tid >> 3, c = tid & 7;
        float acc = 0.f;
        for (int t = 0; t < 8; ++t) acc += R[r * 8 + t] * R[c * 8 + t];
        Rmg[tid] = acc;
    }
    __syncthreads();

    // A powers chain: Apows[bi] = A^{bi+1}
    Apows[tid] = As[tid];
    __syncthreads();
    for (int bi = 1; bi < NH; ++bi) {
        const int r = tid >> 4, c = tid & 15;
        const float* prev = Apows + (bi - 1) * 256;
        float acc = 0.f;
        for (int t = 0; t < 16; ++t) acc += prev[r * 16 + t] * As[t * 16 + c];
        Apows[bi * 256 + tid] = acc;
        __syncthreads();
    }
    // A_hat: linear layout identical to Apows
    for (int idx = tid; idx < NK * NI; idx += NT) Ahat[idx] = Apows[idx];

    // AB blocks: AB[0]=B, AB[i]=A^i B
    if (tid < 128) {
        const int r = tid >> 3, c = tid & 7;
        ABs[tid] = Bs[tid];
        for (int i = 1; i < NH; ++i) {
            const float* Ap = Apows + (i - 1) * 256;
            float acc = 0.f;
            for (int t = 0; t < 16; ++t) acc += Ap[r * 16 + t] * Bs[t * 8 + c];
            ABs[i * 128 + tid] = acc;
        }
    }
    __syncthreads();

    // B_hat (lower block-triangular Toeplitz) and QB = blockdiag(Qm) B_hat
    for (int idx = tid; idx < NK * NU; idx += NT) {
        const int row = idx / NU, col = idx - row * NU;
        const int bi = row >> 4, r = row & 15;
        const int bj = col >> 3, c = col & 7;
        if (bj <= bi) {
            Bhat[idx] = ABs[(bi - bj) * 128 + r * 8 + c];
            float acc = 0.f;
            for (int t = 0; t < 16; ++t)
                acc += Qms[r * 16 + t] * ABs[(bi - bj) * 128 + t * 8 + c];
            QB[idx] = acc;
        } else {
            Bhat[idx] = 0.f;
            QB[idx] = 0.f;
        }
    }
}

// ---------------------------------------------------------------------------
// Kernel 1: Q_hat = B_hat^T QB + kron(I,Rm); twoQ = 2*Q_hat.  WMMA SYRK.
// 45 row-chunk units (15 tile-rows x 3 chunks of 5 column tiles) over 8 waves.
// ---------------------------------------------------------------------------
__global__ void __launch_bounds__(NT) k_qhat(float* __restrict__ ws) {
    const float* Bhat = ws + OFF_BHAT;
    const float* QB   = ws + OFF_QB;
    const float* Rm   = ws + OFF_RM;
    float* Qhat = ws + OFF_QHAT;
    float* twoQ = ws + OFF_TWOQ;
    const int tid = threadIdx.x;
    const int lane = tid & 31, wave = tid >> 5;
    const int half = lane >> 4, lm = lane & 15;

    for (int unit = wave; unit < 45; unit += 8) {
        const int ti = unit / 3, chunk = unit - ti * 3;
        v8f acc[5] = {};
        wmma_syrk480_row5(Bhat + ti * 16 + lm, QB + chunk * 80 + lm,
                          nullptr, half, acc);
#pragma unroll
        for (int u = 0; u < 5; ++u) {
            for (int v = 0; v < 8; ++v) {
                const int r = ti * 16 + v + 8 * half;
                const int c = chunk * 80 + u * 16 + lm;
                float val = acc[u][v];
                if ((r >> 3) == (c >> 3)) val += Rm[(r & 7) * 8 + (c & 7)];
                Qhat[r * NU + c] = val;
                twoQ[r * NU + c] = 2.f * val;
            }
        }
    }
}

// ---------------------------------------------------------------------------
// Kernel 2: per-batch problem data: p, h, and the fixed cost terms b+c.
// One workgroup per batch element.
// ---------------------------------------------------------------------------
__global__ void __launch_bounds__(NT) k_init(const float* __restrict__ x,
                                             const float* __restrict__ s0,
                                             const float* __restrict__ s1,
                                             const float* __restrict__ s2,
                                             float* __restrict__ ws) {
    __shared__ float x0s[16];
    __shared__ float ax[NK];
    __shared__ float red[NT];
    const int b = blockIdx.x, tid = threadIdx.x;
    const float* Ahat = ws + OFF_AHAT;
    const float* QB   = ws + OFF_QB;
    const float* Qm   = ws + OFF_QM;
    float* p  = ws + OFF_P + (size_t)b * NU;
    float* h  = ws + OFF_H + (size_t)b * NINEQ;
    float* bc = ws + OFF_BC;

    if (tid < 16) x0s[tid] = x[b * NI + tid];
    __syncthreads();

    for (int k = tid; k < NK; k += NT) {
        float a = 0.f;
        for (int t = 0; t < 16; ++t) a += Ahat[k * 16 + t] * x0s[t];
        ax[k] = a;
    }
    __syncthreads();

    for (int i = tid; i < NU; i += NT) {
        float a = 0.f;
        for (int k = 0; k < NK; ++k) a += QB[k * NU + i] * ax[k];
        p[i] = 2.f * a;
    }
    for (int j = tid; j < NINEQ; j += NT) {
        float v;
        if (j < 480)       v = s0[j];
        else if (j < 960)  v = s1[j - 480] - ax[j - 480];
        else               v = s2[j - 960] + ax[j - 960];
        h[j] = v;
    }
    // b = sum_blk ax_blk^T Qm ax_blk ; c = x0^T Qm x0
    float part = 0.f;
    for (int k = tid; k < NK; k += NT) {
        const int blk = k >> 4, r = k & 15;
        float y = 0.f;
        for (int t = 0; t < 16; ++t) y += Qm[r * 16 + t] * ax[blk * 16 + t];
        part += ax[k] * y;
    }
    {
        const int r = tid >> 4, c = tid & 15;
        part += x0s[r] * Qm[r * 16 + c] * x0s[c];
    }
    red[tid] = part;
    __syncthreads();
    for (int o = 128; o > 0; o >>= 1) {
        if (tid < o) red[tid] += red[tid + o];
        __syncthreads();
    }
    if (tid == 0) bc[b] = red[0];
}

// ---------------------------------------------------------------------------
// Kernel 3: full 25-iteration PDIP solve + output, one workgroup per batch.
// 240x240 KKT matrix in LDS (231KB of the 320KB WGP LDS); WMMA fp32 SYRK for
// M = 2Q + diag(d1+d2) + B^T diag(e) B; in-LDS Cholesky + tri-solves.
// ---------------------------------------------------------------------------
__global__ void __launch_bounds__(NT) k_pdip(const float* __restrict__ ws,
                                             float* __restrict__ out) {
    extern __shared__ float sm[];
    float* M   = sm + LM_M;     // [240][241]
    float* z   = sm + LM_Z;
    float* rhs = sm + LM_RHS;   // becomes dz after solve
    float* dd  = sm + LM_DD;
    float* Bz  = sm + LM_BZ;    // scratch: Bz / w / Bdz
    float* sqe = sm + LM_SQE;
    float* lam = sm + LM_LAM;
    float* s   = sm + LM_S;
    float* h   = sm + LM_H;
    float* rp  = sm + LM_RP;
    float* t   = sm + LM_T;
    float* dsv = sm + LM_DS;
    float* dlv = sm + LM_DL;
    float* red = sm + LM_RED;

    const int b = blockIdx.x, tid = threadIdx.x;
    const int lane = tid & 31, wave = tid >> 5;
    const int half = lane >> 4, lm = lane & 15;
    const float* Bhat = ws + OFF_BHAT;
    const float* twoQ = ws + OFF_TWOQ;
    const float* Qhat = ws + OFF_QHAT;
    const float* p    = ws + OFF_P + (size_t)b * NU;
    const float* hg   = ws + OFF_H + (size_t)b * NINEQ;

    for (int j = tid; j < NINEQ; j += NT) {
        const float hv = hg[j];
        h[j] = hv;
        s[j] = fmaxf(hv, 1.f);
        lam[j] = 1.f;
    }
    for (int i = tid; i < NU; i += NT) z[i] = 0.f;
    __syncthreads();

    float mu = 0.f;
    for (int it = 0; it < QP_ITERS; ++it) {
        // Bz = B_hat @ z
        for (int k = tid; k < NK; k += NT) {
            const float* row = Bhat + k * NU;
            float a = 0.f;
            for (int i = 0; i < NU; ++i) a += row[i] * z[i];
            Bz[k] = a;
        }
        __syncthreads();
        // r_prim = Gz + s - h; mu = mean(lam*s)
        float msum = 0.f;
        for (int j = tid; j < NINEQ; j += NT) {
            float gz;
            if (j < 240)      gz =  z[j];
            else if (j < 480) gz = -z[j - 240];
            else if (j < 960) gz =  Bz[j - 480];
            else              gz = -Bz[j - 960];
            rp[j] = gz + s[j] - h[j];
            msum += lam[j] * s[j];
        }
        red[tid] = msum;
        __syncthreads();
        for (int o = 128; o > 0; o >>= 1) {
            if (tid < o) red[tid] += red[tid + o];
            __syncthreads();
        }
        mu = red[0] * (1.f / (float)NINEQ);
        __syncthreads();
        // t = (r_cent - lam*r_prim)/s ; e = d3+d4 ; dd = d1+d2
        for (int j = tid; j < NINEQ; j += NT) {
            const float rc = lam[j] * s[j] - SIGMA * mu;
            t[j] = (rc - lam[j] * rp[j]) / s[j];
        }
        for (int k = tid; k < NK; k += NT) {
            const float e = lam[480 + k] / s[480 + k] + lam[960 + k] / s[960 + k];
            sqe[k] = sqrtf(e);
        }
        for (int i = tid; i < NU; i += NT)
            dd[i] = lam[i] / s[i] + lam[240 + i] / s[240 + i];
        __syncthreads();
        // w = (t3-t4) - (lam3-lam4) into Bz scratch
        for (int k = tid; k < NK; k += NT)
            Bz[k] = (t[480 + k] - t[960 + k]) - (lam[480 + k] - lam[960 + k]);
        __syncthreads();
        // rhs = -r_dual + t@G  (fused: -2Qz - p - lam1 + lam2 + t1 - t2 + B^T w)
        for (int i = tid; i < NU; i += NT) {
            const float* row = twoQ + i * NU;
            float tq = 0.f;
            for (int j2 = 0; j2 < NU; ++j2) tq += row[j2] * z[j2];
            float col = 0.f;
            for (int k = 0; k < NK; ++k) col += Bhat[k * NU + i] * Bz[k];
            rhs[i] = -tq - p[i] - lam[i] + lam[240 + i] + t[i] - t[240 + i] + col;
        }
        // M = 2Q + diag(dd) + (sqrt(e) B)^T (sqrt(e) B)   [WMMA fp32 SYRK]
        // 45 row-chunk units (15 tile-rows x 3 chunks) over 8 waves;
        // A-fragment reused across the 5 column tiles of each chunk.
        for (int unit = wave; unit < 45; unit += 8) {
            const int ti = unit / 3, chunk = unit - ti * 3;
            v8f acc[5] = {};
            wmma_syrk480_row5(Bhat + ti * 16 + lm, Bhat + chunk * 80 + lm,
                              sqe, half, acc);
#pragma unroll
            for (int u = 0; u < 5; ++u) {
                for (int v = 0; v < 8; ++v) {
                    const int r = ti * 16 + v + 8 * half;
                    const int c = chunk * 80 + u * 16 + lm;
                    float val = acc[u][v] + twoQ[r * NU + c];
                    if (r == c) val += dd[r];
                    M[r * 241 + c] = val;
                }
            }
        }
        __syncthreads();
        // Cholesky (in LDS): M = L L^T, lower triangle in place
        for (int k = 0; k < NU; ++k) {
            if (tid == 0) M[k * 241 + k] = sqrtf(M[k * 241 + k]);
            __syncthreads();
            const float dk = M[k * 241 + k];
            for (int i = k + 1 + tid; i < NU; i += NT) M[i * 241 + k] /= dk;
            __syncthreads();
            for (int i = k + 1 + tid; i < NU; i += NT) {
                const float lik = M[i * 241 + k];
                for (int j2 = k + 1; j2 <= i; ++j2)
                    M[i * 241 + j2] -= lik * M[j2 * 241 + k];
            }
            __syncthreads();
        }
        // forward solve L y = rhs
        for (int k = 0; k < NU; ++k) {
            if (tid == 0) rhs[k] /= M[k * 241 + k];
            __syncthreads();
            const float yk = rhs[k];
            for (int i = k + 1 + tid; i < NU; i += NT) rhs[i] -= M[i * 241 + k] * yk;
            __syncthreads();
        }
        // backward solve L^T dz = y
        for (int k = NU - 1; k >= 0; --k) {
            if (tid == 0) rhs[k] /= M[k * 241 + k];
            __syncthreads();
            const float xk = rhs[k];
            for (int i = tid; i < k; i += NT) rhs[i] -= M[k * 241 + i] * xk;
            __syncthreads();
        }
        // Bdz into Bz
        for (int k = tid; k < NK; k += NT) {
            const float* row = Bhat + k * NU;
            float a = 0.f;
            for (int i = 0; i < NU; ++i) a += row[i] * rhs[i];
            Bz[k] = a;
        }
        __syncthreads();
        // ds, dlam, fraction-to-boundary step
        float amin = 1e30f;
        for (int j = tid; j < NINEQ; j += NT) {
            float gdz;
            if (j < 240)      gdz =  rhs[j];
            else if (j < 480) gdz = -rhs[j - 240];
            else if (j < 960) gdz =  Bz[j - 480];
            else              gdz = -Bz[j - 960];
            const float dsj = -rp[j] - gdz;
            const float rc = lam[j] * s[j] - SIGMA * mu;
            const float dlj = (-rc - lam[j] * dsj) / s[j];
            dsv[j] = dsj;
            dlv[j] = dlj;
            if (dlj < 0.f) amin = fminf(amin, -lam[j] / dlj);
            if (dsj < 0.f) amin = fminf(amin, -s[j] / dsj);
        }
        red[tid] = amin;
        __syncthreads();
        for (int o = 128; o > 0; o >>= 1) {
            if (tid < o) red[tid] = fminf(red[tid], red[tid + o]);
            __syncthreads();
        }
        const float alpha = fminf(1.f, 0.99f * red[0]);
        __syncthreads();
        for (int i = tid; i < NU; i += NT) z[i] += alpha * rhs[i];
        for (int j = tid; j < NINEQ; j += NT) {
            lam[j] += alpha * dlv[j];
            s[j]   += alpha * dsv[j];
        }
        __syncthreads();
    }

    // cost a = u^T Q_hat u + p.u ; out = [0.1*(a+b+c), u]
    float part = 0.f;
    for (int i = tid; i < NU; i += NT) {
        const float* row = Qhat + i * NU;
        float qz = 0.f;
        for (int j2 = 0; j2 < NU; ++j2) qz += row[j2] * z[j2];
        part += z[i] * (qz + p[i]);
    }
    red[tid] = part;
    __syncthreads();
    for (int o = 128; o > 0; o >>= 1) {
        if (tid < o) red[tid] += red[tid + o];
        __syncthreads();
    }
    if (tid == 0) out[(size_t)b * 241] = 0.1f * (red[0] + ws[OFF_BC + b]);
    for (int i = tid; i < NU; i += NT) out[(size_t)b * 241 + 1 + i] = z[i];
}

// ---------------------------------------------------------------------------
extern "C" void kernel_launch(void* const* d_in, const int* in_sizes, int n_in,
                              void* d_out, int out_size, void* d_ws, size_t ws_size,
                              hipStream_t stream) {
    (void)in_sizes; (void)n_in; (void)out_size; (void)ws_size;
    const float* x  = (const float*)d_in[0];
    const float* Q  = (const float*)d_in[1];
    const float* R  = (const float*)d_in[2];
    const float* A  = (const float*)d_in[3];
    const float* B  = (const float*)d_in[4];
    const float* s0 = (const float*)d_in[5];
    const float* s1 = (const float*)d_in[6];
    const float* s2 = (const float*)d_in[7];
    float* out = (float*)d_out;
    float* ws  = (float*)d_ws;

    (void)hipFuncSetAttribute((const void*)k_pdip,
                              hipFuncAttributeMaxDynamicSharedMemorySize,
                              PDIP_SMEM_BYTES);

    k_precompute<<<1, NT, 0, stream>>>(Q, R, A, B, ws);
    k_qhat<<<1, NT, 0, stream>>>(ws);
    k_init<<<BATCH, NT, 0, stream>>>(x, s0, s1, s2, ws);
    k_pdip<<<BATCH, NT, PDIP_SMEM_BYTES, stream>>>(ws, out);
}